// VICRegL_10934986736203
// MI455X (gfx1250) — compile-verified
//
#include <hip/hip_runtime.h>
#include <hip/hip_bf16.h>
#include <math.h>

// ---------------------------------------------------------------------------
// VICRegL local loss for MI455X (gfx1250, wave32, WMMA).
// B=4096 batches, L=49 rows, C=128 features, D=2 locations.
// Matmul-shaped work (per-batch Gram matrices for cdist, per-m 128x128
// covariance X^T X over 4096 samples) runs on v_wmma_f32_16x16x4_f32.
// ---------------------------------------------------------------------------

typedef float v2f __attribute__((ext_vector_type(2)));
typedef float v8f __attribute__((ext_vector_type(8)));

#define NB    4096
#define NL    49
#define NC    128
#define LPAD  132   // LDS row stride for k_nn tiles (conflict-free frag reads)
#define CPAD  136   // LDS row stride for k_cov chunks
#define DPAD  52    // LDS row stride for distance matrices

// 8 vicreg calls v=0..7:
//   v0 featRow k=20 x=m1 | v1 featCol k=4  x=m2 | v2 locRow k=20 x=m1
//   v3 locCol  k=4  x=m2 | v4 featCol k=20 x=m2 | v5 featRow k=4 x=m1
//   v6 locCol  k=20 x=m2 | v7 locRow  k=4  x=m1
__constant__ int c_ks[8]   = {20, 4, 20, 4, 20, 4, 20, 4};
__constant__ int c_off[8]  = {0, 20, 24, 44, 48, 68, 72, 92};   // m-slot prefix (total 96)
__constant__ int c_xsrc[8] = {0, 1, 0, 1, 1, 0, 1, 0};          // 0 -> maps_1, 1 -> maps_2

__global__ void k_init(float* acc) {
    if (threadIdx.x < 8) acc[threadIdx.x] = 0.0f;
}

// ---------------------------------------------------------------------------
// Kernel 2: per-batch distance matrices (feature via WMMA Gram, location
// direct), then row/col nearest neighbors for both. 128 threads = 4 waves.
// ---------------------------------------------------------------------------
__global__ void k_nn(const float* __restrict__ m1, const float* __restrict__ m2,
                     const float* __restrict__ l1, const float* __restrict__ l2,
                     float* __restrict__ nnval, int* __restrict__ nnidx) {
    extern __shared__ float sm[];
    float* sA   = sm;                      // 64 x LPAD
    float* sB   = sA + 64 * LPAD;          // 64 x LPAD
    float* nA   = sB + 64 * LPAD;          // 64
    float* nB   = nA + 64;                 // 64
    float* sdF  = nB + 64;                 // 49 x DPAD (squared feat dist)
    float* sdL  = sdF + NL * DPAD;         // 49 x DPAD (squared loc dist)
    float* sloc = sdL + NL * DPAD;         // 49*2 (loc1) + 49*2 (loc2)

    const int b   = blockIdx.x;
    const int tid = threadIdx.x;

    // Stage both 49x128 tiles (rows 49..63 zero-padded for WMMA).
    for (int idx = tid; idx < 64 * NC; idx += 128) {
        int r = idx >> 7, c = idx & 127;
        float va = 0.0f, vb = 0.0f;
        if (r < NL) {
            va = m1[((size_t)b * NL + r) * NC + c];
            vb = m2[((size_t)b * NL + r) * NC + c];
        }
        sA[r * LPAD + c] = va;
        sB[r * LPAD + c] = vb;
    }
    for (int idx = tid; idx < NL * 2; idx += 128) {
        sloc[idx]          = l1[(size_t)b * (NL * 2) + idx];
        sloc[NL * 2 + idx] = l2[(size_t)b * (NL * 2) + idx];
    }
    __syncthreads();

    // Row norms.
    if (tid < NL) {
        float a = 0.0f, bb = 0.0f;
        for (int c = 0; c < NC; ++c) {
            float x = sA[tid * LPAD + c]; a  += x * x;
            float y = sB[tid * LPAD + c]; bb += y * y;
        }
        nA[tid] = a; nB[tid] = bb;
    }
    __syncthreads();

    // Gram matrix G = m1 * m2^T via f32 WMMA; wave w owns row tile i0 = 16*w.
    const int lane = tid & 31;
    const int wave = tid >> 5;
    const int i0   = wave * 16;
    const int ml   = lane & 15;
    const int kh   = (lane >> 4) << 1;        // 0 or 2 (K half-select)
    const int ih   = (lane >> 4) * 8;         // C/D row-half offset

    for (int j0 = 0; j0 < 64; j0 += 16) {
        v8f ct = {};
        for (int k = 0; k < NC; k += 4) {
            int kb = k + kh;
            v2f a, bf;
            a.x  = sA[(i0 + ml) * LPAD + kb];
            a.y  = sA[(i0 + ml) * LPAD + kb + 1];
            bf.x = sB[(j0 + ml) * LPAD + kb];
            bf.y = sB[(j0 + ml) * LPAD + kb + 1];
            ct = __builtin_amdgcn_wmma_f32_16x16x4_f32(
                    false, a, false, bf, (short)0, ct, false, false);
        }
        int j = j0 + ml;
        for (int r = 0; r < 8; ++r) {
            int i = i0 + ih + r;
            if (i < NL && j < NL) {
                float d2 = nA[i] + nB[j] - 2.0f * ct[r];
                sdF[i * DPAD + j] = fmaxf(d2, 0.0f);  // sqrt skipped: rank-invariant
            }
        }
    }
    __syncthreads();

    // Location squared distances (D=2).
    for (int idx = tid; idx < NL * NL; idx += 128) {
        int i = idx / NL, j = idx % NL;
        float dx = sloc[i * 2 + 0] - sloc[NL * 2 + j * 2 + 0];
        float dy = sloc[i * 2 + 1] - sloc[NL * 2 + j * 2 + 1];
        sdL[i * DPAD + j] = dx * dx + dy * dy;
    }
    __syncthreads();

    // NN scans: src 0=featRow, 1=featCol, 2=locRow, 3=locCol. First-min on ties.
    if (tid < NL) {
        float bv; int bi;
        bv = sdF[tid * DPAD]; bi = 0;
        for (int j = 1; j < NL; ++j) { float x = sdF[tid * DPAD + j]; if (x < bv) { bv = x; bi = j; } }
        nnval[((size_t)0 * NB + b) * NL + tid] = bv;
        nnidx[((size_t)0 * NB + b) * NL + tid] = bi;

        bv = sdF[tid]; bi = 0;
        for (int i = 1; i < NL; ++i) { float x = sdF[i * DPAD + tid]; if (x < bv) { bv = x; bi = i; } }
        nnval[((size_t)1 * NB + b) * NL + tid] = bv;
        nnidx[((size_t)1 * NB + b) * NL + tid] = bi;

        bv = sdL[tid * DPAD]; bi = 0;
        for (int j = 1; j < NL; ++j) { float x = sdL[tid * DPAD + j]; if (x < bv) { bv = x; bi = j; } }
        nnval[((size_t)2 * NB + b) * NL + tid] = bv;
        nnidx[((size_t)2 * NB + b) * NL + tid] = bi;

        bv = sdL[tid]; bi = 0;
        for (int i = 1; i < NL; ++i) { float x = sdL[i * DPAD + tid]; if (x < bv) { bv = x; bi = i; } }
        nnval[((size_t)3 * NB + b) * NL + tid] = bv;
        nnidx[((size_t)3 * NB + b) * NL + tid] = bi;
    }
}

// ---------------------------------------------------------------------------
// Kernel 3: stable top-k selection == sort(argsort(minvals)[:k]).
// rank(t) = #{ j : v[j] < v[t] or (v[j]==v[t] and j<t) }; keep rank<k in
// row-index order.
// ---------------------------------------------------------------------------
__global__ void k_select(const float* __restrict__ nnval, const int* __restrict__ nnidx,
                         int* __restrict__ selb, int* __restrict__ nnib) {
    __shared__ float v[NL];
    __shared__ int   id[NL];
    __shared__ int   rk[NL];
    const int b = blockIdx.x, t = threadIdx.x;
    const int cfg20[4] = {0, 4, 2, 6};
    const int cfg4[4]  = {5, 1, 7, 3};

    for (int src = 0; src < 4; ++src) {
        __syncthreads();
        if (t < NL) {
            v[t]  = nnval[((size_t)src * NB + b) * NL + t];
            id[t] = nnidx[((size_t)src * NB + b) * NL + t];
        }
        __syncthreads();
        if (t < NL) {
            float mv = v[t]; int r = 0;
            for (int j = 0; j < NL; ++j) {
                float vj = v[j];
                if (vj < mv || (vj == mv && j < t)) r++;
            }
            rk[t] = r;
        }
        __syncthreads();
        if (t < NL) {
            int r = rk[t];
            if (r < 20) {
                int pos = 0;
                for (int j = 0; j < t; ++j) if (rk[j] < 20) pos++;
                int cfg = cfg20[src];
                selb[((size_t)cfg * NB + b) * 20 + pos] = t;
                nnib[((size_t)cfg * NB + b) * 20 + pos] = id[t];
            }
            if (r < 4) {
                int pos = 0;
                for (int j = 0; j < t; ++j) if (rk[j] < 4) pos++;
                int cfg = cfg4[src];
                selb[((size_t)cfg * NB + b) * 20 + pos] = t;
                nnib[((size_t)cfg * NB + b) * 20 + pos] = id[t];
            }
        }
    }
}

// ---------------------------------------------------------------------------
// Kernel 4: per (call v, match m): column sums of x and y over the batch, and
// the invariance term sum((x-y)^2) (pre-scaled into the accumulator).
// ---------------------------------------------------------------------------
__global__ void k_sums(const float* __restrict__ m1, const float* __restrict__ m2,
                       const int* __restrict__ selb, const int* __restrict__ nnib,
                       float* __restrict__ sumx, float* __restrict__ sumy,
                       float* __restrict__ acc) {
    const int g = blockIdx.x;                       // 0..95
    int v = 0;
    while (v < 7 && g >= c_off[v + 1]) v++;
    const int m = g - c_off[v];
    const int M = c_ks[v];
    const float* mx = c_xsrc[v] ? m2 : m1;
    const float* my = c_xsrc[v] ? m1 : m2;
    const int c = threadIdx.x;                      // 0..127

    float sx = 0.0f, sy = 0.0f, rp = 0.0f;
    for (int b = 0; b < NB; ++b) {
        int sr = selb[((size_t)v * NB + b) * 20 + m];
        int nr = nnib[((size_t)v * NB + b) * 20 + m];
        float x = mx[((size_t)b * NL + sr) * NC + c];
        float y = my[((size_t)b * NL + nr) * NC + c];
        sx += x; sy += y;
        float d = x - y; rp += d * d;
    }
    sumx[g * NC + c] = sx;
    sumy[g * NC + c] = sy;

    __shared__ float red[128];
    red[c] = rp;
    __syncthreads();
    for (int s = 64; s > 0; s >>= 1) { if (c < s) red[c] += red[c + s]; __syncthreads(); }
    if (c == 0) {
        float scale = 0.25f * 25.0f / ((float)NB * (float)M * (float)NC);
        atomicAdd(acc, red[0] * scale);
    }
}

// ---------------------------------------------------------------------------
// Kernel 5: per (call v, match m, side): S = X^T X over 4096 gathered rows
// via f32 WMMA, then cov = (S - sc*sd/B)/(B-1); diag -> std/relu term,
// off-diag -> cov^2 term; pre-scaled atomicAdd into the loss accumulator.
// 256 threads = 8 waves; wave w owns output tile row c0 = 16*w (8 tiles).
// ---------------------------------------------------------------------------
__global__ void k_cov(const float* __restrict__ m1, const float* __restrict__ m2,
                      const int* __restrict__ selb, const int* __restrict__ nnib,
                      const float* __restrict__ sumx, const float* __restrict__ sumy,
                      float* __restrict__ acc_g) {
    extern __shared__ float sm[];
    float* chunk = sm;                 // 32 x CPAD
    float* red   = sm + 32 * CPAD;     // 256

    const int g2   = blockIdx.x;       // 0..191
    const int g    = g2 >> 1;
    const int side = g2 & 1;
    int v = 0;
    while (v < 7 && g >= c_off[v + 1]) v++;
    const int m = g - c_off[v];
    const int M = c_ks[v];
    const float* src  = (side == 0) ? (c_xsrc[v] ? m2 : m1)
                                    : (c_xsrc[v] ? m1 : m2);
    const int*   rows = (side == 0) ? selb : nnib;
    const float* sums = (side == 0) ? sumx : sumy;

    const int tid  = threadIdx.x;
    const int lane = tid & 31;
    const int wave = tid >> 5;
    const int ml   = lane & 15;
    const int kh   = (lane >> 4) << 1;
    const int ih   = (lane >> 4) * 8;
    const int c0   = wave * 16;

    v8f ct[8] = {};

    for (int bc = 0; bc < NB / 32; ++bc) {        // 128 chunks of 32 batches
        const int b0 = bc * 32;
        __syncthreads();
        for (int idx = tid; idx < 32 * NC; idx += 256) {
            int r = idx >> 7, c = idx & 127;
            int b = b0 + r;
            int sr = rows[((size_t)v * NB + b) * 20 + m];
            chunk[r * CPAD + c] = src[((size_t)b * NL + sr) * NC + c];
        }
        __syncthreads();
        for (int k = 0; k < 32; k += 4) {
            int kb = k + kh;
            v2f a;
            a.x = chunk[kb * CPAD + c0 + ml];
            a.y = chunk[(kb + 1) * CPAD + c0 + ml];
            #pragma unroll
            for (int t = 0; t < 8; ++t) {
                int d0 = t * 16;
                v2f bf;
                bf.x = chunk[kb * CPAD + d0 + ml];
                bf.y = chunk[(kb + 1) * CPAD + d0 + ml];
                ct[t] = __builtin_amdgcn_wmma_f32_16x16x4_f32(
                            false, a, false, bf, (short)0, ct[t], false, false);
            }
        }
    }

    // Epilogue: each lane's ct[t][r] is S[c0+ih+r][t*16+ml].
    const float invB   = 1.0f / (float)NB;
    const float invBm1 = 1.0f / (float)(NB - 1);
    float stdsum = 0.0f, covsum = 0.0f;
    for (int t = 0; t < 8; ++t) {
        int d = t * 16 + ml;
        float sud = sums[g * NC + d];
        for (int r = 0; r < 8; ++r) {
            int ci = c0 + ih + r;
            float suc = sums[g * NC + ci];
            float cov = (ct[t][r] - suc * sud * invB) * invBm1;
            if (ci == d) {
                float stdv = sqrtf(cov + 1e-4f);
                stdsum += fmaxf(1.0f - stdv, 0.0f);
            } else {
                covsum += cov * cov;
            }
        }
    }
    __syncthreads();
    float mc = (float)M * (float)NC;
    red[tid] = stdsum * (0.25f * 25.0f * 0.5f / mc)
             + covsum * (0.25f * 1.0f * 0.5f / mc);
    __syncthreads();
    for (int s = 128; s > 0; s >>= 1) { if (tid < s) red[tid] += red[tid + s]; __syncthreads(); }
    if (tid == 0) atomicAdd(acc_g, red[0]);
}

__global__ void k_fin(const float* __restrict__ acc, float* __restrict__ out) {
    if (threadIdx.x == 0 && blockIdx.x == 0) out[0] = acc[0];
}

// ---------------------------------------------------------------------------
extern "C" void kernel_launch(void* const* d_in, const int* in_sizes, int n_in,
                              void* d_out, int out_size, void* d_ws, size_t ws_size,
                              hipStream_t stream) {
    (void)in_sizes; (void)n_in; (void)out_size; (void)ws_size;
    const float* m1 = (const float*)d_in[0];
    const float* m2 = (const float*)d_in[1];
    const float* l1 = (const float*)d_in[2];
    const float* l2 = (const float*)d_in[3];

    char* w = (char*)d_ws;
    size_t o = 0;
    float* nnval = (float*)(w + o); o += (size_t)4 * NB * NL * sizeof(float);
    int*   nnidx = (int*)  (w + o); o += (size_t)4 * NB * NL * sizeof(int);
    int*   selb  = (int*)  (w + o); o += (size_t)8 * NB * 20 * sizeof(int);
    int*   nnib  = (int*)  (w + o); o += (size_t)8 * NB * 20 * sizeof(int);
    float* sumx  = (float*)(w + o); o += (size_t)96 * NC * sizeof(float);
    float* sumy  = (float*)(w + o); o += (size_t)96 * NC * sizeof(float);
    float* acc   = (float*)(w + o); o += 16 * sizeof(float);

    k_init<<<1, 64, 0, stream>>>(acc);

    size_t smem_nn = (size_t)(64 * LPAD * 2 + 64 * 2 + NL * DPAD * 2 + NL * 4) * sizeof(float);
    k_nn<<<NB, 128, smem_nn, stream>>>(m1, m2, l1, l2, nnval, nnidx);

    k_select<<<NB, 64, 0, stream>>>(nnval, nnidx, selb, nnib);

    k_sums<<<96, 128, 0, stream>>>(m1, m2, selb, nnib, sumx, sumy, acc);

    size_t smem_cov = (size_t)(32 * CPAD + 256) * sizeof(float);
    k_cov<<<192, 256, smem_cov, stream>>>(m1, m2, selb, nnib, sumx, sumy, acc);

    k_fin<<<1, 32, 0, stream>>>(acc, (float*)d_out);
}